// regression_model_82927228551821
// MI455X (gfx1250) — compile-verified
//
#include <hip/hip_runtime.h>
#include <math.h>

typedef __attribute__((ext_vector_type(2))) float v2f;
typedef __attribute__((ext_vector_type(8))) float v8f;

#define BLOCK 128

// --- LDS layout (all offsets in floats, kept even for 8B-aligned float2 loads) ---
constexpr int BSZ     = 1508;   // per-branch weight block
constexpr int W0T_OFS = 0;      // W0t[32][10] : W0t[n][k] = W0[k][n], zero-padded
constexpr int W1T_OFS = 320;    // W1t[32][34] : W1t[n][k] = W1[k][n], zero-padded
constexpr int B0_OFS  = 1408;   // [32] zero-padded
constexpr int B1_OFS  = 1440;   // [32] zero-padded
constexpr int W2_OFS  = 1472;   // [32] zero-padded
constexpr int B2_OFS  = 1504;   // [1]
constexpr int HB_STRIDE   = 34; // padded activation-tile row stride (bank-conflict-free)
constexpr int HBUF_OFS    = 7 * BSZ;                          // 10556
constexpr int SMEM_FLOATS = HBUF_OFS + 4 * 32 * HB_STRIDE;    // 14908 floats = 59632 B

struct Params { const float* p[42]; };

__device__ __forceinline__ float fast_tanh(float x) {
#if __has_builtin(__builtin_amdgcn_tanhf)
  return __builtin_amdgcn_tanhf(x);          // gfx1250 v_tanh_f32
#elif __has_builtin(__builtin_amdgcn_tanh_f32)
  return __builtin_amdgcn_tanh_f32(x);
#else
  return tanhf(x);
#endif
}

__device__ __forceinline__ v8f wmma4(v2f a, v2f b, v8f c) {
  // V_WMMA_F32_16X16X4_F32 : D = A(16x4) * B(4x16) + C, full f32
  return __builtin_amdgcn_wmma_f32_16x16x4_f32(false, a, false, b, (short)0, c,
                                               false, false);
}

// One MLP branch for the 32 rows handled by this wave.
// aS0[m] / aS1[m]: pre-built layer-0 A fragments (K-step 0 and 1) for M-tile m.
// Returns this lane's row scalar:  (tanh(tanh(x W0 + b0) W1 + b1)) W2 + b2
template <int DIN>
__device__ __forceinline__ float branch_eval(const v2f aS0[2], const v2f aS1[2],
                                             const float* __restrict__ wb,
                                             float* __restrict__ hbuf, int lane) {
  const int  l15  = lane & 15;
  const bool hi   = lane >= 16;
  const int  kofs = hi ? 2 : 0;           // A/B lane K-offset per ISA 16x4 layout
  constexpr int KS0 = (DIN + 3) / 4;      // K-steps for layer 0 (1 or 2)

  // ---------- layer 0 : h0 = tanh(X * W0 + b0) ----------
  #pragma unroll
  for (int m = 0; m < 2; ++m) {
    v8f c[2];
    #pragma unroll
    for (int n = 0; n < 2; ++n) {
      const float bv = wb[B0_OFS + n * 16 + l15];
      v8f cc = {bv, bv, bv, bv, bv, bv, bv, bv};
      #pragma unroll
      for (int s = 0; s < KS0; ++s) {
        const float2 w = *(const float2*)&wb[W0T_OFS + (n * 16 + l15) * 10 + 4 * s + kofs];
        v2f bf = {w.x, w.y};
        cc = wmma4(s == 0 ? aS0[m] : aS1[m], bf, cc);
      }
      c[n] = cc;
    }
    #pragma unroll
    for (int n = 0; n < 2; ++n)
      #pragma unroll
      for (int r = 0; r < 8; ++r) {
        const int rr = m * 16 + r + (hi ? 8 : 0);       // C layout: VGPR r -> M=r / r+8
        hbuf[rr * HB_STRIDE + n * 16 + l15] = fast_tanh(c[n][r]);
      }
  }

  // ---------- layer 1 : h1 = tanh(h0 * W1 + b1) ----------
  // Per M-tile: rows of m=0 / m=1 are disjoint, so only the within-m
  // read-h0 / write-h1 overlap matters -> preload this m's 8 A fragments.
  #pragma unroll
  for (int m = 0; m < 2; ++m) {
    v2f a1[8];
    #pragma unroll
    for (int s = 0; s < 8; ++s) {
      const float2 h = *(const float2*)&hbuf[(m * 16 + l15) * HB_STRIDE + 4 * s + kofs];
      v2f a = {h.x, h.y};
      a1[s] = a;
    }
    v8f c[2];
    #pragma unroll
    for (int n = 0; n < 2; ++n) {
      const float bv = wb[B1_OFS + n * 16 + l15];
      v8f cc = {bv, bv, bv, bv, bv, bv, bv, bv};
      #pragma unroll
      for (int s = 0; s < 8; ++s) {
        const float2 w = *(const float2*)&wb[W1T_OFS + (n * 16 + l15) * HB_STRIDE + 4 * s + kofs];
        v2f bf = {w.x, w.y};
        cc = wmma4(a1[s], bf, cc);
      }
      c[n] = cc;
    }
    #pragma unroll
    for (int n = 0; n < 2; ++n)
      #pragma unroll
      for (int r = 0; r < 8; ++r) {
        const int rr = m * 16 + r + (hi ? 8 : 0);
        hbuf[rr * HB_STRIDE + n * 16 + l15] = fast_tanh(c[n][r]);
      }
  }

  // ---------- layer 2 : per-lane dot product over own row ----------
  float acc = wb[B2_OFS];
  #pragma unroll
  for (int n = 0; n < 16; ++n) {
    const float2 h = *(const float2*)&hbuf[lane * HB_STRIDE + 2 * n];
    acc = fmaf(h.x, wb[W2_OFS + 2 * n], acc);
    acc = fmaf(h.y, wb[W2_OFS + 2 * n + 1], acc);
  }
  return acc;
}

__global__ __launch_bounds__(BLOCK)
__attribute__((amdgpu_waves_per_eu(4)))
void mlp7_fused_kernel(const float* __restrict__ X, float* __restrict__ out,
                       Params P, int nrows) {
  __shared__ __align__(16) float smem[SMEM_FLOATS];
  const int tid = threadIdx.x;

  // ---- cooperative weight staging: transpose + zero-pad into LDS ----
  constexpr int DIN[7] = {3, 3, 6, 6, 3, 3, 3};  // alpha,x0,Dz,Dy,yv,gamma,h
  for (int b = 0; b < 7; ++b) {
    const float* W0g = P.p[6 * b + 0];
    const float* b0g = P.p[6 * b + 1];
    const float* W1g = P.p[6 * b + 2];
    const float* b1g = P.p[6 * b + 3];
    const float* W2g = P.p[6 * b + 4];
    const float* b2g = P.p[6 * b + 5];
    float* base = smem + b * BSZ;
    const int d = DIN[b];
    for (int i = tid; i < 320; i += BLOCK) {            // W0t[32][10]
      const int n = i / 10, k = i % 10;
      base[W0T_OFS + i] = (n < 30 && k < d) ? W0g[k * 30 + n] : 0.f;
    }
    for (int i = tid; i < 32 * HB_STRIDE; i += BLOCK) { // W1t[32][34]
      const int n = i / HB_STRIDE, k = i % HB_STRIDE;
      base[W1T_OFS + i] = (n < 30 && k < 30) ? W1g[k * 30 + n] : 0.f;
    }
    if (tid < 32) {
      base[B0_OFS + tid] = tid < 30 ? b0g[tid] : 0.f;
      base[B1_OFS + tid] = tid < 30 ? b1g[tid] : 0.f;
      base[W2_OFS + tid] = tid < 30 ? W2g[tid] : 0.f;
    }
    if (tid == 0) base[B2_OFS] = b2g[0];
  }
  __syncthreads();

  const int lane    = tid & 31;
  const int l15     = lane & 15;
  const bool hi     = lane >= 16;
  const int kofs    = hi ? 2 : 0;
  const int wid     = tid >> 5;
  const int rowBase = blockIdx.x * BLOCK + wid * 32;
  const int rowMax  = nrows - 1;
  float* hbuf = smem + HBUF_OFS + wid * (32 * HB_STRIDE);

  // ---- layer-0 A fragments, shared by all 7 branches (A operand is just X) ----
  // din=3, s=0: lanes<16 (X0,X1), lanes>=16 (X2, 0)
  // din=6, s=0: lanes<16 (X0,X1), lanes>=16 (X2,X3)
  // din=6, s=1: lanes<16 (X4,X5), lanes>=16 (0, 0)
  v2f a3[2], a6s0[2], a6s1[2];
  #pragma unroll
  for (int m = 0; m < 2; ++m) {
    int r = rowBase + m * 16 + l15;
    if (r > rowMax) r = rowMax;
    const float v0 = X[r * 6 + kofs];
    const float v1 = X[r * 6 + kofs + 1];
    v2f t6 = {v0, v1};                  a6s0[m] = t6;
    v2f t3 = {v0, hi ? 0.f : v1};       a3[m]   = t3;
    v2f t1 = {hi ? 0.f : X[r * 6 + 4],
              hi ? 0.f : X[r * 6 + 5]}; a6s1[m] = t1;
  }

  float br[7];
  br[0] = branch_eval<3>(a3,   a6s1, smem + 0 * BSZ, hbuf, lane); // alpha
  br[1] = branch_eval<3>(a3,   a6s1, smem + 1 * BSZ, hbuf, lane); // x0
  br[2] = branch_eval<6>(a6s0, a6s1, smem + 2 * BSZ, hbuf, lane); // Dz
  br[3] = branch_eval<6>(a6s0, a6s1, smem + 3 * BSZ, hbuf, lane); // Dy
  br[4] = branch_eval<3>(a3,   a6s1, smem + 4 * BSZ, hbuf, lane); // yv
  br[5] = branch_eval<3>(a3,   a6s1, smem + 5 * BSZ, hbuf, lane); // gamma
  br[6] = branch_eval<3>(a3,   a6s1, smem + 6 * BSZ, hbuf, lane); // h

  // ---- fused f32 epilogue (this lane's own row) ----
  int row = rowBase + lane;
  const int srow = row > rowMax ? rowMax : row;
  const float X0 = X[srow * 6 + 0];
  const float X1 = X[srow * 6 + 1];
  const float X3 = X[srow * 6 + 3];
  const float X4 = X[srow * 6 + 4];
  const float X5 = X[srow * 6 + 5];

  const float chi = X3 / X0;
  const float yi  = X4 / X0;
  const float zi  = X5 / X0;

  const float alpha = br[0];
  const float x0v   = br[1];
  const float Dz    = expf(br[2]);
  const float Dy    = expf(br[3]);
  const float yv    = expf(br[4]);
  const float gamma = br[5];
  const float sh    = expf(br[6]);

  const float dchi = chi - x0v;
  const float lamz = fabsf(sqrtf(Dz * dchi)) + 1e-7f;
  const float lamy = fabsf(sqrtf(Dy * dchi)) + 1e-7f;
  const float ksi  = zi / lamz;
  const float eta  = yi / lamy;
  const float g    = 0.5f * ksi * expf(-0.25f * ksi * ksi);
  const float hg   = 0.28209479177387814f * expf(-0.25f * eta * eta); // 0.5/sqrt(pi)
  const float r0l  = X0 / lamz;
  const float f    = alpha * (X1 / lamy) * r0l * r0l * g * hg;

  const float yp   = yv - yi;
  const float al   = yp * yp + sh * sh + zi * zi;
  const float tzh  = 2.f * zi * sh;
  const float den  = al * al - tzh * tzh + 1e-7f;
  const float fprime = fminf(gamma * sh * yp * chi / den, 2.0f);

  if (row <= rowMax) out[row] = expf(f + fprime);
}

extern "C" void kernel_launch(void* const* d_in, const int* in_sizes, int n_in,
                              void* d_out, int out_size, void* d_ws, size_t ws_size,
                              hipStream_t stream) {
  (void)n_in; (void)d_ws; (void)ws_size; (void)out_size;
  const float* X = (const float*)d_in[0];
  Params P;
  for (int i = 0; i < 42; ++i) P.p[i] = (const float*)d_in[1 + i];
  const int nrows = in_sizes[0] / 6;              // 2,000,000
  const int grid  = (nrows + BLOCK - 1) / BLOCK;  // 15625, exact
  mlp7_fused_kernel<<<grid, BLOCK, 0, stream>>>(X, (float*)d_out, P, nrows);
}